// VoronoiNoise_59906203844814
// MI455X (gfx1250) — compile-verified
//
#include <hip/hip_runtime.h>

typedef __attribute__((ext_vector_type(2))) float v2f;
typedef __attribute__((ext_vector_type(8))) float v8f;

#define VW      512
#define VH      512
#define VB      2
#define NPTS    289
#define NTILES  19            // ceil(289/16) point tiles
#define GROUPS  8             // 16-pixel groups handled per wave
#define HW      (VW * VH)
#define GROUPS_PER_IMG (HW / 16)   // 16384 (divisible by GROUPS -> one batch per wave)

// ---------------------------------------------------------------------------
// Kernel 0: init per-image min/max slots (lo = +inf, hi = 0; distances >= 0)
// ---------------------------------------------------------------------------
__global__ void voro_init(float* slots) {
    int t = threadIdx.x;
    if (t < VB)           slots[t] = __int_as_float(0x7F800000);  // lo[b] = +inf
    else if (t < 2 * VB)  slots[t] = 0.0f;                        // hi[b] = 0
}

// ---------------------------------------------------------------------------
// Kernel 1: per-pixel nearest-point distance via V_WMMA_F32_16X16X4_F32.
//   dist^2 = |p|^2 + (-2 p.q + |q|^2);  the parenthesized term is a 16x16
//   block product of A(points)[16x4] x B(pixels)[4x16].
//   All 19 point tiles stay resident in VGPRs for the whole wave.
// ---------------------------------------------------------------------------
__global__ __launch_bounds__(256)
void voro_main(const float* __restrict__ pts,   // [B, 289, 2]
               float* __restrict__ mind,        // [B, H*W]
               float* __restrict__ slots) {     // [lo0, lo1, hi0, hi1]
    const int lane  = threadIdx.x & 31;
    const int m     = lane & 15;
    const bool lo16 = lane < 16;
    const int wave  = (blockIdx.x * blockDim.x + threadIdx.x) >> 5;
    const int g0    = wave * GROUPS;                 // first 16-pixel group
    const int b     = g0 / GROUPS_PER_IMG;           // whole wave in one image

    // --- Preload all point tiles as WMMA A-matrices (reused for every group).
    // A-matrix 16x4 f32 layout: lanes 0-15 hold K=0,1 ; lanes 16-31 hold K=2,3.
    //   row m (point m): [-2qx, -2qy, |q|^2, 0]
    v2f A[NTILES];
    const float* pb = pts + b * NPTS * 2;
    #pragma unroll
    for (int t = 0; t < NTILES; ++t) {
        int  pi    = t * 16 + m;
        bool valid = pi < NPTS;
        float qx = 0.f, qy = 0.f;
        if (valid) { qx = pb[pi * 2 + 0]; qy = pb[pi * 2 + 1]; }
        float a0, a1;
        if (lo16) {                       // K=0,1 : -2qx, -2qy
            a0 = valid ? -2.0f * qx : 0.0f;
            a1 = valid ? -2.0f * qy : 0.0f;
        } else {                          // K=2,3 : |q|^2, 0 (pad -> huge)
            a0 = valid ? (qx * qx + qy * qy) : 3.0e37f;
            a1 = 0.0f;
        }
        A[t][0] = a0;
        A[t][1] = a1;
    }

    float wlo = __int_as_float(0x7F800000);
    float whi = 0.0f;

    for (int gi = 0; gi < GROUPS; ++gi) {
        int g      = g0 + gi;
        int local  = g - b * GROUPS_PER_IMG;
        int pixel  = local * 16 + m;              // this lane's pixel (n = m)
        float px   = (float)(pixel & (VW - 1));
        float py   = (float)(pixel >> 9);

        // B-matrix 4x16 f32: column n (pixel n) = [px, py, 1, 0]
        v2f Bm;
        Bm[0] = lo16 ? px : 1.0f;
        Bm[1] = lo16 ? py : 0.0f;

        v8f run;
        #pragma unroll
        for (int r = 0; r < 8; ++r) run[r] = 3.0e37f;
        v8f cz = {};

        #pragma unroll
        for (int t = 0; t < NTILES; ++t) {
            // D[m_point, n_pixel] = -2 p_n . q_m + |q_m|^2
            v8f d = __builtin_amdgcn_wmma_f32_16x16x4_f32(
                false, A[t], false, Bm, (short)0, cz, false, false);
            #pragma unroll
            for (int r = 0; r < 8; ++r) run[r] = fminf(run[r], d[r]);
        }

        // Min over rows: 8 VGPRs (M=0..7 / 8..15) then half-wave combine.
        float s = run[0];
        #pragma unroll
        for (int r = 1; r < 8; ++r) s = fminf(s, run[r]);
        s = fminf(s, __shfl_xor(s, 16, 32));

        float d2   = s + px * px + py * py;
        float dist = sqrtf(fmaxf(d2, 0.0f));

        if (lo16) mind[b * HW + pixel] = dist;
        wlo = fminf(wlo, dist);
        whi = fmaxf(whi, dist);
    }

    // Wave-level reduce, then one pair of atomics per wave.
    #pragma unroll
    for (int off = 16; off >= 1; off >>= 1) {
        wlo = fminf(wlo, __shfl_xor(wlo, off, 32));
        whi = fmaxf(whi, __shfl_xor(whi, off, 32));
    }
    if (lane == 0) {
        // Non-negative IEEE floats: int-bit order == float order.
        atomicMin((int*)&slots[b],      __float_as_int(wlo));
        atomicMax((int*)&slots[VB + b], __float_as_int(whi));
    }
}

// ---------------------------------------------------------------------------
// Kernel 2: per-image min-max normalize, replicate to 3 channels.
// ---------------------------------------------------------------------------
__global__ void voro_norm(const float* __restrict__ mind,
                          const float* __restrict__ slots,
                          float* __restrict__ out) {
    int i = blockIdx.x * blockDim.x + threadIdx.x;
    if (i >= VB * HW) return;
    int b = i / HW;
    float lo = slots[b];
    float hi = slots[VB + b];
    float n  = (mind[i] - lo) / (hi - lo);
    out[i * 3 + 0] = n;
    out[i * 3 + 1] = n;
    out[i * 3 + 2] = n;
}

// ---------------------------------------------------------------------------
extern "C" void kernel_launch(void* const* d_in, const int* in_sizes, int n_in,
                              void* d_out, int out_size, void* d_ws, size_t ws_size,
                              hipStream_t stream) {
    const float* pts = (const float*)d_in[0];   // [B, 289, 2] float32
    float* out   = (float*)d_out;               // [B, H, W, 3] float32
    float* mind  = (float*)d_ws;                // B*H*W floats scratch
    float* slots = mind + VB * HW;              // 4 floats: lo[2], hi[2]

    voro_init<<<1, 32, 0, stream>>>(slots);

    const int total_waves = (VB * HW / 16) / GROUPS;   // 4096 waves
    const int waves_per_block = 8;                     // 256 threads
    voro_main<<<total_waves / waves_per_block, 256, 0, stream>>>(pts, mind, slots);

    voro_norm<<<(VB * HW + 255) / 256, 256, 0, stream>>>(mind, slots, out);
}